// LHGNets_17566416241059
// MI455X (gfx1250) — compile-verified
//
#include <hip/hip_runtime.h>
#include <hip/hip_bf16.h>
#include <math.h>

// Problem constants (from reference): N=2048, D=128, V=3, H=4, VD=512, K=10
#define NN 2048
#define DD 128
#define VV 3
#define HH 4
#define VDIM 512
#define KNNK 10
#define VH (VV * HH)
#define META_LD (VV * HH * DD)   // 1536
#define ALPHA_LRELU 0.2f

typedef __attribute__((ext_vector_type(2))) float v2f;
typedef __attribute__((ext_vector_type(8))) float v8f;

// ---------------------------------------------------------------------------
// fp32 WMMA: D(16x16,f32) = A(16x4,f32) * B(4x16,f32) + C
// A layout: lane(0..15)=row M, half(lane>>4) selects K-pair; vgpr = K&1
// B layout: lane(0..15)=col N, half selects K-pair; vgpr = K&1
// C/D layout: vgpr v, lanes 0-15 -> (M=v, N=lane); lanes 16-31 -> (M=v+8)
// ---------------------------------------------------------------------------
__device__ __forceinline__ v8f wmma_f32_4(v2f a, v2f b, v8f c) {
  return __builtin_amdgcn_wmma_f32_16x16x4_f32(
      /*neg_a=*/false, a, /*neg_b=*/false, b,
      /*c_mod=*/(short)0, c, /*reuse_a=*/false, /*reuse_b=*/false);
}

// One wave computes a 16x64 tile of C = A*B (+bias)(+elu).
// A row-major [M,Kdim] lda; B row-major [Kdim,Ncols] ldb; C row-major ldc.
__device__ __forceinline__ void gemm_tile16x64(
    const float* __restrict__ A, const float* __restrict__ B,
    const float* __restrict__ bias, float* __restrict__ C,
    int Kdim, int lda, int ldb, int ldc, int m0, int n0, int act) {
  const int lane = threadIdx.x & 31;
  const int r    = lane & 15;
  const int half = lane >> 4;
  const int ko   = half << 1;

  v8f acc[4] = {};
  const float* arow = A + (size_t)(m0 + r) * lda + ko;
  const float* bcol = B + n0 + r;

  for (int k0 = 0; k0 < Kdim; k0 += 4) {
    v2f a = *(const v2f*)(arow + k0);
    const float* bk = bcol + (size_t)(k0 + ko) * ldb;
#pragma unroll
    for (int t = 0; t < 4; ++t) {
      v2f b;
      b.x = bk[t * 16];
      b.y = bk[t * 16 + ldb];
      acc[t] = wmma_f32_4(a, b, acc[t]);
    }
  }
#pragma unroll
  for (int t = 0; t < 4; ++t) {
#pragma unroll
    for (int vv = 0; vv < 8; ++vv) {
      int row = m0 + vv + (half << 3);
      int col = n0 + t * 16 + r;
      float val = acc[t][vv];
      if (bias) val += bias[col];
      if (act == 1) val = (val > 0.f) ? val : (__expf(val) - 1.f);  // elu
      C[(size_t)row * ldc + col] = val;
    }
  }
}

// ---------------------------------------------------------------------------
// 1) per-row squared norms of latent
// ---------------------------------------------------------------------------
__global__ __launch_bounds__(256) void sqnorm_kernel(const float* __restrict__ X,
                                                     float* __restrict__ sq) {
  const int wave = threadIdx.x >> 5, lane = threadIdx.x & 31;
  const int n = blockIdx.x * 8 + wave;
  const float* row = X + (size_t)n * DD;
  float s = 0.f;
#pragma unroll
  for (int i = 0; i < 4; ++i) { float w = row[lane + 32 * i]; s += w * w; }
#pragma unroll
  for (int m = 16; m; m >>= 1) s += __shfl_xor(s, m, 32);
  if (lane == 0) sq[n] = s;
}

// ---------------------------------------------------------------------------
// 2) dist[i,j] = sq[i] + sq[j] - 2 * <x_i, x_j>   via fp32 WMMA (B = X^T)
// ---------------------------------------------------------------------------
__global__ __launch_bounds__(32) void gram_dist_kernel(const float* __restrict__ X,
                                                       const float* __restrict__ sq,
                                                       float* __restrict__ dist) {
  const int lane = threadIdx.x & 31;
  const int r = lane & 15, half = lane >> 4, ko = half << 1;
  const int m0 = blockIdx.x * 16;
  const int n0 = blockIdx.y * 64;

  v8f acc[4] = {};
  const float* arow = X + (size_t)(m0 + r) * DD + ko;
  const float* brow = X + (size_t)(n0 + r) * DD + ko;  // B[k][n] = X[n][k]

  for (int k0 = 0; k0 < DD; k0 += 4) {
    v2f a = *(const v2f*)(arow + k0);
#pragma unroll
    for (int t = 0; t < 4; ++t) {
      v2f b = *(const v2f*)(brow + (size_t)t * 16 * DD + k0);
      acc[t] = wmma_f32_4(a, b, acc[t]);
    }
  }
#pragma unroll
  for (int t = 0; t < 4; ++t) {
#pragma unroll
    for (int vv = 0; vv < 8; ++vv) {
      int row = m0 + vv + (half << 3);
      int col = n0 + t * 16 + r;
      dist[(size_t)row * NN + col] = sq[row] + sq[col] - 2.f * acc[t][vv];
    }
  }
}

// ---------------------------------------------------------------------------
// 3) top-10 nearest per row (smallest distance, ties -> lower index)
// ---------------------------------------------------------------------------
__global__ __launch_bounds__(256) void topk_kernel(const float* __restrict__ dist,
                                                   int* __restrict__ knn) {
  __shared__ float sd[NN];
  __shared__ float rv[256];
  __shared__ int   ri[256];
  const int n = blockIdx.x, t = threadIdx.x;
  for (int j = t; j < NN; j += 256) sd[j] = dist[(size_t)n * NN + j];
  __syncthreads();
  for (int k = 0; k < KNNK; ++k) {
    float best = 3.0e38f; int bi = NN;
    for (int j = t; j < NN; j += 256) {
      float v = sd[j];
      if (v < best) { best = v; bi = j; }   // ascending scan keeps lowest index
    }
    rv[t] = best; ri[t] = bi;
    __syncthreads();
    for (int s = 128; s > 0; s >>= 1) {
      if (t < s) {
        if (rv[t + s] < rv[t] || (rv[t + s] == rv[t] && ri[t + s] < ri[t])) {
          rv[t] = rv[t + s]; ri[t] = ri[t + s];
        }
      }
      __syncthreads();
    }
    if (t == 0) { knn[n * 16 + k] = ri[0]; sd[ri[0]] = 3.0e38f; }
    __syncthreads();
  }
}

// ---------------------------------------------------------------------------
// 4) rec[v] = latent @ enc_W[v] + enc_b[v]
// ---------------------------------------------------------------------------
__global__ __launch_bounds__(32) void rec_gemm_kernel(const float* __restrict__ latent,
                                                      const float* __restrict__ encW,
                                                      const float* __restrict__ encb,
                                                      float* __restrict__ rec) {
  const int v = blockIdx.z;
  gemm_tile16x64(latent, encW + (size_t)v * DD * VDIM, encb + (size_t)v * VDIM,
                 rec + (size_t)v * NN * VDIM, DD, DD, VDIM, VDIM,
                 blockIdx.x * 16, blockIdx.y * 64, 0);
}

// ---------------------------------------------------------------------------
// 5) Wh[vh] = latent @ gat_W[vh]
// ---------------------------------------------------------------------------
__global__ __launch_bounds__(32) void wh_gemm_kernel(const float* __restrict__ latent,
                                                     const float* __restrict__ gatW,
                                                     float* __restrict__ Wh) {
  const int vh = blockIdx.z;
  gemm_tile16x64(latent, gatW + (size_t)vh * DD * DD, nullptr,
                 Wh + (size_t)vh * NN * DD, DD, DD, DD, DD,
                 blockIdx.x * 16, blockIdx.y * 64, 0);
}

// ---------------------------------------------------------------------------
// 6) f1/f2 = Wh . a1 / Wh . a2 (wave-per-row dot products)
// ---------------------------------------------------------------------------
__global__ __launch_bounds__(256) void f12_kernel(const float* __restrict__ Wh,
                                                  const float* __restrict__ gata,
                                                  float* __restrict__ f1,
                                                  float* __restrict__ f2) {
  const int vh = blockIdx.y;
  const int wave = threadIdx.x >> 5, lane = threadIdx.x & 31;
  const int n = blockIdx.x * 8 + wave;
  const float* row = Wh + ((size_t)vh * NN + n) * DD;
  const float* a1 = gata + (size_t)vh * 2 * DD;
  const float* a2 = a1 + DD;
  float s1 = 0.f, s2 = 0.f;
#pragma unroll
  for (int i = 0; i < 4; ++i) {
    float w = row[lane + 32 * i];
    s1 += w * a1[lane + 32 * i];
    s2 += w * a2[lane + 32 * i];
  }
#pragma unroll
  for (int m = 16; m; m >>= 1) { s1 += __shfl_xor(s1, m, 32); s2 += __shfl_xor(s2, m, 32); }
  if (lane == 0) { f1[(size_t)vh * NN + n] = s1; f2[(size_t)vh * NN + n] = s2; }
}

// ---------------------------------------------------------------------------
// 7) Sparse GAT: softmax over <=11 neighbors (10 KNN + self), gather-accumulate.
//    Exactly matches dense masked-softmax: exp(NEG - max) underflows to 0.
//    meta[n, vh*128 + d] = sum_j att_j * Wh[vh, j, d]
// ---------------------------------------------------------------------------
__global__ __launch_bounds__(32) void gat_attn_kernel(const float* __restrict__ Wh,
                                                      const float* __restrict__ f1,
                                                      const float* __restrict__ f2,
                                                      const int* __restrict__ knn,
                                                      const float* __restrict__ fmask,
                                                      float* __restrict__ meta) {
  const int n = blockIdx.x, vh = blockIdx.y;
  const int v = vh >> 2;  // H = 4
  const int lane = threadIdx.x & 31;

  const float mn = fmask[(size_t)n * VV + v];
  int j = n;
  bool valid = (lane == KNNK);  // self slot, always unmasked (eye term)
  if (lane < KNNK) {
    int jj = knn[n * 16 + lane];
    j = jj;
    valid = (jj != n) && (mn > 0.f) && (fmask[(size_t)jj * VV + v] > 0.f);
  }

  float e = -__builtin_inff();
  if (valid) {
    float x = f1[(size_t)vh * NN + n] + f2[(size_t)vh * NN + j];
    e = (x > 0.f) ? x : ALPHA_LRELU * x;  // leaky_relu
  }
  float emax = e;
#pragma unroll
  for (int m = 16; m; m >>= 1) emax = fmaxf(emax, __shfl_xor(emax, m, 32));
  float p = valid ? __expf(e - emax) : 0.f;
  float psum = p;
#pragma unroll
  for (int m = 16; m; m >>= 1) psum += __shfl_xor(psum, m, 32);
  const float att = p / psum;

  float a0 = 0.f, a1 = 0.f, a2 = 0.f, a3 = 0.f;
  for (int t = 0; t <= KNNK; ++t) {
    float at = __shfl(att, t, 32);
    int jt = __shfl(j, t, 32);
    if (at != 0.f) {  // wave-uniform branch
      const float* wr = Wh + ((size_t)vh * NN + jt) * DD;
      a0 += at * wr[lane];
      a1 += at * wr[lane + 32];
      a2 += at * wr[lane + 64];
      a3 += at * wr[lane + 96];
    }
  }
  float* mrow = meta + (size_t)n * META_LD + (size_t)vh * DD;
  mrow[lane] = a0; mrow[lane + 32] = a1; mrow[lane + 64] = a2; mrow[lane + 96] = a3;
}

// ---------------------------------------------------------------------------
// 8) semantic = meta @ agg_W + agg_b        (K = 1536)
// 9) logits   = elu(semantic @ cls_W + cls_b)
// ---------------------------------------------------------------------------
__global__ __launch_bounds__(32) void agg_gemm_kernel(const float* __restrict__ meta,
                                                      const float* __restrict__ aggW,
                                                      const float* __restrict__ aggb,
                                                      float* __restrict__ semantic) {
  gemm_tile16x64(meta, aggW, aggb, semantic, META_LD, META_LD, DD, DD,
                 blockIdx.x * 16, blockIdx.y * 64, 0);
}

__global__ __launch_bounds__(32) void cls_gemm_kernel(const float* __restrict__ semantic,
                                                      const float* __restrict__ clsW,
                                                      const float* __restrict__ clsb,
                                                      float* __restrict__ logits) {
  gemm_tile16x64(semantic, clsW, clsb, logits, DD, DD, DD, DD,
                 blockIdx.x * 16, blockIdx.y * 64, 1 /*elu*/);
}

// ---------------------------------------------------------------------------
// 10) output = log_softmax(logits, axis=1)   (wave per row, 4 cols/lane)
// ---------------------------------------------------------------------------
__global__ __launch_bounds__(256) void lsm_kernel(const float* __restrict__ logits,
                                                  float* __restrict__ outp) {
  const int wave = threadIdx.x >> 5, lane = threadIdx.x & 31;
  const int n = blockIdx.x * 8 + wave;
  const float* row = logits + (size_t)n * DD;
  float x0 = row[lane], x1 = row[lane + 32], x2 = row[lane + 64], x3 = row[lane + 96];
  float mx = fmaxf(fmaxf(x0, x1), fmaxf(x2, x3));
#pragma unroll
  for (int m = 16; m; m >>= 1) mx = fmaxf(mx, __shfl_xor(mx, m, 32));
  float s = __expf(x0 - mx) + __expf(x1 - mx) + __expf(x2 - mx) + __expf(x3 - mx);
#pragma unroll
  for (int m = 16; m; m >>= 1) s += __shfl_xor(s, m, 32);
  const float lse = mx + __logf(s);
  float* orow = outp + (size_t)n * DD;
  orow[lane] = x0 - lse; orow[lane + 32] = x1 - lse;
  orow[lane + 64] = x2 - lse; orow[lane + 96] = x3 - lse;
}

// ---------------------------------------------------------------------------
extern "C" void kernel_launch(void* const* d_in, const int* in_sizes, int n_in,
                              void* d_out, int out_size, void* d_ws, size_t ws_size,
                              hipStream_t stream) {
  (void)in_sizes; (void)n_in; (void)out_size; (void)ws_size;

  const float* fmask  = (const float*)d_in[0];  // [N,V]
  const float* latent = (const float*)d_in[1];  // [N,D]
  const float* encW   = (const float*)d_in[2];  // [V,D,VD]
  const float* encb   = (const float*)d_in[3];  // [V,VD]
  const float* gatW   = (const float*)d_in[4];  // [V,H,D,D]
  const float* gata   = (const float*)d_in[5];  // [V,H,2D]
  const float* aggW   = (const float*)d_in[6];  // [V*H*D, D]
  const float* aggb   = (const float*)d_in[7];  // [D]
  const float* clsW   = (const float*)d_in[8];  // [D,D]
  const float* clsb   = (const float*)d_in[9];  // [D]

  // Outputs: (rec [V,N,VD], output [N,D], semantic [N,D]) concatenated flat
  float* out      = (float*)d_out;
  float* rec      = out;
  float* outp     = out + (size_t)VV * NN * VDIM;
  float* semantic = outp + (size_t)NN * DD;

  // Workspace layout (floats). dist is dead after topk -> meta/logits alias it.
  float* ws   = (float*)d_ws;
  float* sq   = ws;                                   // 2048
  int*   knn  = (int*)(ws + NN);                      // 2048*16 ints
  float* f1   = ws + NN + NN * 16;                    // 12*2048
  float* f2   = f1 + (size_t)VH * NN;                 // 12*2048
  float* Wh   = f2 + (size_t)VH * NN;                 // 12*2048*128
  float* dist = Wh + (size_t)VH * NN * DD;            // 2048*2048
  float* meta = dist;                                 // 2048*1536 (alias)
  float* logits = dist + (size_t)NN * META_LD;        // 2048*128  (alias)

  // 1) row norms
  sqnorm_kernel<<<NN / 8, 256, 0, stream>>>(latent, sq);
  // 2) pairwise squared distances (WMMA fp32)
  gram_dist_kernel<<<dim3(NN / 16, NN / 64), 32, 0, stream>>>(latent, sq, dist);
  // 3) KNN top-10 per row
  topk_kernel<<<NN, 256, 0, stream>>>(dist, knn);
  // 4) per-view decoders (WMMA fp32)
  rec_gemm_kernel<<<dim3(NN / 16, VDIM / 64, VV), 32, 0, stream>>>(latent, encW, encb, rec);
  // 5) Wh projections (WMMA fp32)
  wh_gemm_kernel<<<dim3(NN / 16, DD / 64, VH), 32, 0, stream>>>(latent, gatW, Wh);
  // 6) attention logit halves
  f12_kernel<<<dim3(NN / 8, VH), 256, 0, stream>>>(Wh, gata, f1, f2);
  // 7) sparse masked-softmax attention + neighbor aggregation -> meta
  gat_attn_kernel<<<dim3(NN, VH), 32, 0, stream>>>(Wh, f1, f2, knn, fmask, meta);
  // 8) semantic = meta @ agg_W + agg_b (WMMA fp32, K=1536)
  agg_gemm_kernel<<<dim3(NN / 16, DD / 64), 32, 0, stream>>>(meta, aggW, aggb, semantic);
  // 9) logits = elu(semantic @ cls_W + cls_b) (WMMA fp32)
  cls_gemm_kernel<<<dim3(NN / 16, DD / 64), 32, 0, stream>>>(semantic, clsW, clsb, logits);
  // 10) output = log_softmax(logits)
  lsm_kernel<<<NN / 8, 256, 0, stream>>>(logits, outp);
}